// QLayer_13898514170710
// MI455X (gfx1250) — compile-verified
//
#include <hip/hip_runtime.h>
#include <math.h>

#define DIM   256
#define BATCH 32768
#define KTOT  512

typedef float v2f __attribute__((ext_vector_type(2)));
typedef float v8f __attribute__((ext_vector_type(8)));

// ---------------------------------------------------------------------------
// 1) Build scaled generators Xs (squeezing) and Xd (displacement), and the
//    first Horner term P = I + X/T.  X is pre-scaled by 1/2^s (s=6).
// ---------------------------------------------------------------------------
__global__ void build_gen_kernel(const float* __restrict__ r_p,
                                 const float* __restrict__ bx_p,
                                 const float* __restrict__ bp_p,
                                 float2* __restrict__ X,   // 2 matrices
                                 float2* __restrict__ P,   // 2 matrices
                                 float invT, float inv2s) {
    int j = threadIdx.x;
    int i = blockIdx.x;
    float r  = r_p[0];
    float bx = bx_p[0];
    float bp = bp_p[0];

    // squeezing generator: 0.5*(conj(r)*a*a - r*adag*adag), r real
    float2 xs; xs.x = 0.f; xs.y = 0.f;
    if (j == i + 2) xs.x =  0.5f * r * sqrtf((float)((i + 1) * (i + 2)));
    if (j == i - 2) xs.x = -0.5f * r * sqrtf((float)(i * (i - 1)));
    xs.x *= inv2s;

    // displacement generator: alpha*adag - conj(alpha)*a, alpha = bx + i*bp
    float2 xd; xd.x = 0.f; xd.y = 0.f;
    if (j == i - 1) { float s = sqrtf((float)i);       xd.x =  bx * s; xd.y = bp * s; }
    if (j == i + 1) { float s = sqrtf((float)(i + 1)); xd.x = -bx * s; xd.y = bp * s; }
    xd.x *= inv2s; xd.y *= inv2s;

    int idx = i * DIM + j;
    X[idx]                 = xs;
    X[idx + DIM * DIM]     = xd;
    float id = (i == j) ? 1.f : 0.f;
    float2 ps; ps.x = id + xs.x * invT; ps.y = xs.y * invT;
    float2 pd; pd.x = id + xd.x * invT; pd.y = xd.y * invT;
    P[idx]             = ps;
    P[idx + DIM * DIM] = pd;
}

// ---------------------------------------------------------------------------
// 2) Dual-batch 256x256 complex matmul: Out[m] = (addI? I:0) + scale*A[m]*B[m]
//    blockIdx.y selects matrix m.  Row per block, column per thread.
// ---------------------------------------------------------------------------
__global__ void cmatmul_kernel(const float2* __restrict__ A,
                               const float2* __restrict__ B,
                               float2* __restrict__ Out,
                               float scale, int addI) {
    int j = threadIdx.x;
    int i = blockIdx.x;
    int m = blockIdx.y;
    const float2* Am = A + (size_t)m * DIM * DIM;
    const float2* Bm = B + (size_t)m * DIM * DIM;
    float2*       Om = Out + (size_t)m * DIM * DIM;

    float accr = 0.f, acci = 0.f;
    for (int k = 0; k < DIM; ++k) {
        float2 a = Am[i * DIM + k];   // broadcast across block
        float2 b = Bm[k * DIM + j];   // coalesced
        accr = fmaf(a.x, b.x, fmaf(-a.y, b.y, accr));
        acci = fmaf(a.x, b.y, fmaf( a.y, b.x, acci));
    }
    accr *= scale; acci *= scale;
    if (addI && i == j) accr += 1.f;
    float2 o; o.x = accr; o.y = acci;
    Om[i * DIM + j] = o;
}

// ---------------------------------------------------------------------------
// 3) Fold rotations: S'[i][j] = e^{-i*th2*i} * S[i][j] * e^{-i*th1*j}
// ---------------------------------------------------------------------------
__global__ void rot_scale_kernel(float2* __restrict__ S,
                                 const float* __restrict__ t1_p,
                                 const float* __restrict__ t2_p) {
    int j = threadIdx.x;
    int i = blockIdx.x;
    double ph = -((double)t2_p[0] * (double)i + (double)t1_p[0] * (double)j);
    float c = (float)cos(ph), s = (float)sin(ph);
    float2 v = S[i * DIM + j];
    float2 o; o.x = v.x * c - v.y * s; o.y = v.x * s + v.y * c;
    S[i * DIM + j] = o;
}

// ---------------------------------------------------------------------------
// 4) Build the transposed real operator Wt[k][n] (512x512) from U0 with the
//    Kerr phase e^{i*kappa*i^2} folded in per output row i.
//    n<256: real-part column (row i=n):   Wt = [ Re(U) | -Im(U) ]
//    n>=256: imag-part column (i=n-256):  Wt = [ Im(U) |  Re(U) ]
// ---------------------------------------------------------------------------
__global__ void build_wt_kernel(const float2* __restrict__ U0,
                                const float* __restrict__ kap_p,
                                float* __restrict__ Wt) {
    int n0 = threadIdx.x;        // 0..255
    int nh = blockIdx.y;         // 0 = real column, 1 = imag column
    int k  = blockIdx.x;         // 0..511
    int i  = n0;                 // output row of U
    int j  = k & 255;

    double ph = (double)kap_p[0] * (double)i * (double)i;
    float c = (float)cos(ph), s = (float)sin(ph);
    float2 u0 = U0[i * DIM + j];
    float ur = u0.x * c - u0.y * s;
    float ui = u0.x * s + u0.y * c;

    float val;
    if (nh == 0) val = (k < DIM) ? ur : -ui;
    else         val = (k < DIM) ? ui :  ur;
    Wt[(size_t)k * KTOT + nh * DIM + n0] = val;
}

// ---------------------------------------------------------------------------
// 5) Main batched apply as one real GEMM using V_WMMA_F32_16X16X4_F32.
//    C(32768 x 512) = A'(32768 x 512) * Wt, A' = [Xr | Xi] (virtual).
//    Block: 256 threads = 8 wave32s, 2(M) x 4(N) waves.
//    Wave tile: M=64 (4 subtiles) x {real,imag} column pair (2 subtiles).
//    Block tile: M=128, 64 complex output columns.  Grid: 256 x 4.
// ---------------------------------------------------------------------------
#define LDA 20   // padded LDS row pitch (floats) -> conflict-free strided reads

__global__ __launch_bounds__(256)
void qgemm_kernel(const float* __restrict__ Xr, const float* __restrict__ Xi,
                  const float* __restrict__ Wt, float* __restrict__ out) {
    __shared__ float shA[128 * LDA];

    int tid   = threadIdx.x;
    int lane  = tid & 31;
    int w     = tid >> 5;
    int waveM = w & 1;           // 0..1
    int waveN = w >> 1;          // 0..3
    int lm    = lane & 15;       // N / M index within subtile
    int hi    = lane >> 4;       // K-pair select (A/B operand layout)

    int bm      = blockIdx.x * 128;            // batch-row base
    int colbase = blockIdx.y * 64 + waveN * 16;// complex column base (0..255)

    v8f acc[4][2];
    for (int mi = 0; mi < 4; ++mi)
        for (int ni = 0; ni < 2; ++ni)
            acc[mi][ni] = (v8f){0.f,0.f,0.f,0.f,0.f,0.f,0.f,0.f};

    for (int k0 = 0; k0 < KTOT; k0 += 16) {
        // ---- stage A' tile [128 rows x 16 k] into LDS (coalesced float4) ----
        const float* src = (k0 < DIM) ? Xr : Xi;
        int kk0 = k0 & 255;
        __syncthreads();
        #pragma unroll
        for (int i = 0; i < 2; ++i) {
            int lin = tid + i * 256;       // 0..511 float4 chunks
            int row = lin >> 2;            // 0..127
            int seg = lin & 3;             // 0..3
            const float4 vv =
                *(const float4*)(src + (size_t)(bm + row) * DIM + kk0 + seg * 4);
            float* dst = &shA[row * LDA + seg * 4];
            dst[0] = vv.x; dst[1] = vv.y; dst[2] = vv.z; dst[3] = vv.w;
        }
        __syncthreads();

        #pragma unroll
        for (int kk = 0; kk < 16; kk += 4) {
            int kb = kk + 2 * hi;
            // A operands from LDS: lane holds A[row][kb], A[row][kb+1]
            v2f a[4];
            #pragma unroll
            for (int mi = 0; mi < 4; ++mi) {
                int row = waveM * 64 + mi * 16 + lm;
                float2 t = *(const float2*)&shA[row * LDA + kb];
                a[mi].x = t.x; a[mi].y = t.y;
            }
            // B operands from global Wt (coalesced b32 per half-wave)
            int kg = k0 + kb;
            v2f b[2];
            #pragma unroll
            for (int ni = 0; ni < 2; ++ni) {
                int col = colbase + ni * DIM + lm;   // real block / imag block
                b[ni].x = Wt[(size_t)kg * KTOT + col];
                b[ni].y = Wt[(size_t)(kg + 1) * KTOT + col];
            }
            #pragma unroll
            for (int mi = 0; mi < 4; ++mi)
                #pragma unroll
                for (int ni = 0; ni < 2; ++ni)
                    acc[mi][ni] = __builtin_amdgcn_wmma_f32_16x16x4_f32(
                        false, a[mi], false, b[ni], (short)0, acc[mi][ni],
                        false, false);
        }
    }

    // ---- epilogue: interleave (re, im) -> coalesced float2 complex stores ----
    int n = colbase + lm;                 // complex column 0..255
    #pragma unroll
    for (int mi = 0; mi < 4; ++mi) {
        int mbase = bm + waveM * 64 + mi * 16 + hi * 8;
        #pragma unroll
        for (int rr = 0; rr < 8; ++rr) {
            float2 v; v.x = acc[mi][0][rr]; v.y = acc[mi][1][rr];
            *(float2*)&out[(size_t)(mbase + rr) * KTOT + 2 * n] = v;
        }
    }
}

// ---------------------------------------------------------------------------
extern "C" void kernel_launch(void* const* d_in, const int* in_sizes, int n_in,
                              void* d_out, int out_size, void* d_ws, size_t ws_size,
                              hipStream_t stream) {
    (void)in_sizes; (void)n_in; (void)out_size; (void)ws_size;
    const float* xr = (const float*)d_in[0];
    const float* xi = (const float*)d_in[1];
    const float* t1 = (const float*)d_in[2];
    const float* t2 = (const float*)d_in[3];
    const float* rr = (const float*)d_in[4];
    const float* bx = (const float*)d_in[5];
    const float* bp = (const float*)d_in[6];
    const float* kp = (const float*)d_in[7];

    char* ws = (char*)d_ws;
    const size_t MS = (size_t)DIM * DIM * sizeof(float2);   // 512 KB
    float2* X  = (float2*)(ws);            // 2 matrices
    float2* P  = (float2*)(ws + 2 * MS);   // 2 matrices (ping)
    float2* Q  = (float2*)(ws + 4 * MS);   // 2 matrices (pong)
    float2* U  = (float2*)(ws + 6 * MS);   // 1 matrix
    float*  Wt = (float*)(ws + 7 * MS);    // 512x512 floats (1 MB)

    const int TORD = 12;   // Taylor order (||X||/64 <~ 0.6 even at 3 sigma)
    const int SQN  = 6;    // squarings: scale = 1/2^6

    build_gen_kernel<<<dim3(DIM), DIM, 0, stream>>>(
        rr, bx, bp, X, P, 1.f / (float)TORD, 1.f / 64.f);

    // Horner: P = I + X/T ; then P <- I + (X*P)/k for k = T-1..1
    float2 *pp = P, *qq = Q;
    for (int k = TORD - 1; k >= 1; --k) {
        cmatmul_kernel<<<dim3(DIM, 2), DIM, 0, stream>>>(X, pp, qq, 1.f / (float)k, 1);
        float2* t = pp; pp = qq; qq = t;
    }
    // squaring phase: P <- P*P, six times
    for (int t = 0; t < SQN; ++t) {
        cmatmul_kernel<<<dim3(DIM, 2), DIM, 0, stream>>>(pp, pp, qq, 1.f, 0);
        float2* tm = pp; pp = qq; qq = tm;
    }
    // pp[0] = S (squeezing), pp[1] = D (displacement)

    rot_scale_kernel<<<dim3(DIM), DIM, 0, stream>>>(pp, t1, t2);        // S' = R2*S*R1
    cmatmul_kernel<<<dim3(DIM, 1), DIM, 0, stream>>>(
        pp + DIM * DIM, pp, U, 1.f, 0);                                  // U0 = D*S'
    build_wt_kernel<<<dim3(KTOT, 2), DIM, 0, stream>>>(U, kp, Wt);       // + Kerr

    qgemm_kernel<<<dim3(BATCH / 128, 4), 256, 0, stream>>>(
        xr, xi, Wt, (float*)d_out);
}